// MDTA_87454124082206
// MI455X (gfx1250) — compile-verified
//
#include <hip/hip_runtime.h>
#include <hip/hip_bf16.h>

// ---------------------------------------------------------------------------
// MDTA (Restormer transposed attention) for gfx1250 / MI455X.
// All GEMM stages use V_WMMA_F32_16X16X4_F32 (full f32 precision; the
// problem is HBM-bound at ~600MB traffic, so f32 WMMA is the right point).
// ---------------------------------------------------------------------------

typedef __attribute__((ext_vector_type(2))) float v2f;
typedef __attribute__((ext_vector_type(8))) float v8f;

#define BATCH 8
#define HH    128
#define WW    128
#define CC    192
#define C3    576
#define HWSZ  16384        // 128*128
#define HEADS 4
#define CHEAD 48
#define SLEN  16384        // hw
#define PAIRS (BATCH*HEADS)          // 32
#define ROWS_PER_BATCH (HEADS*CHEAD) // 192
#define QBUF_PER_BATCH (HWSZ*CC)     // 3145728

__device__ __forceinline__ v8f wmma_f32_16x16x4(v2f a, v2f b, v8f c) {
  // 8 args: (neg_a, A, neg_b, B, c_mod, C, reuse_a, reuse_b)
  return __builtin_amdgcn_wmma_f32_16x16x4_f32(
      false, a, false, b, (short)0, c, false, false);
}

// ---------------------------------------------------------------------------
// K1: qkv = X[131072,192] @ W[192,576]  (one wave -> 16x64 output tile)
// ---------------------------------------------------------------------------
__global__ __launch_bounds__(256) void k_qkv_gemm(
    const float* __restrict__ X, const float* __restrict__ W,
    float* __restrict__ QKV) {
  const int lane = threadIdx.x & 31;
  const int wave = threadIdx.x >> 5;
  const int job  = blockIdx.x * 8 + wave;   // 8192 mtiles * 9 nsupers = 73728
  const int mtile = job / 9;
  const int n0    = (job % 9) * 64;
  const int li    = lane & 15;
  const int koff  = (lane >> 4) << 1;       // 0 for lanes 0-15, 2 for 16-31
  const float* arow = X + (size_t)(mtile * 16 + li) * CC;
  v8f acc0 = {}, acc1 = {}, acc2 = {}, acc3 = {};
  #pragma unroll 4
  for (int k = 0; k < CC; k += 4) {
    v2f a; a.x = arow[k + koff]; a.y = arow[k + koff + 1];
    const float* b0 = W + (size_t)(k + koff) * C3 + n0 + li;
    const float* b1 = b0 + C3;
    v2f b;
    b.x = b0[0];  b.y = b1[0];  acc0 = wmma_f32_16x16x4(a, b, acc0);
    b.x = b0[16]; b.y = b1[16]; acc1 = wmma_f32_16x16x4(a, b, acc1);
    b.x = b0[32]; b.y = b1[32]; acc2 = wmma_f32_16x16x4(a, b, acc2);
    b.x = b0[48]; b.y = b1[48]; acc3 = wmma_f32_16x16x4(a, b, acc3);
  }
  const int rbase = mtile * 16 + ((lane >> 4) << 3);
  float* out = QKV + n0 + li;
  #pragma unroll
  for (int j = 0; j < 8; ++j) {
    size_t r = (size_t)(rbase + j) * C3;
    out[r +  0] = acc0[j];
    out[r + 16] = acc1[j];
    out[r + 32] = acc2[j];
    out[r + 48] = acc3[j];
  }
}

// ---------------------------------------------------------------------------
// K2: depthwise 3x3 SAME over qkv, split into contiguous q/k/v buffers that
// realize the raw reshape: per batch, flat [hw*192 + c] == [192,16384] rows.
// ---------------------------------------------------------------------------
__global__ __launch_bounds__(256) void k_dwconv_split(
    const float* __restrict__ QKV, const float* __restrict__ DW,
    float* __restrict__ Q, float* __restrict__ K, float* __restrict__ V) {
  int idx = blockIdx.x * 256 + threadIdx.x;  // B*128*128*144 = 18874368
  int c4 = idx % 144; int t = idx / 144;
  int x  = t % WW;    t /= WW;
  int y  = t % HH;    t /= HH;
  int b  = t;
  const int ch = c4 * 4;
  float4 acc = make_float4(0.f, 0.f, 0.f, 0.f);
  #pragma unroll
  for (int dy = -1; dy <= 1; ++dy) {
    int yy = y + dy;
    if ((unsigned)yy >= (unsigned)HH) continue;
    #pragma unroll
    for (int dx = -1; dx <= 1; ++dx) {
      int xx = x + dx;
      if ((unsigned)xx >= (unsigned)WW) continue;
      const float4 v = *(const float4*)(QKV + ((size_t)(b * HWSZ + yy * WW + xx) * C3 + ch));
      const float4 w = *(const float4*)(DW + (size_t)((dy + 1) * 3 + (dx + 1)) * C3 + ch);
      acc.x += v.x * w.x; acc.y += v.y * w.y;
      acc.z += v.z * w.z; acc.w += v.w * w.w;
    }
  }
  const int part = ch / CC;       // 0=q, 1=k, 2=v (4-channel group never splits)
  const int ci   = ch % CC;
  float* dst = (part == 0) ? Q : ((part == 1) ? K : V);
  *(float4*)(dst + (size_t)b * QBUF_PER_BATCH + (size_t)(y * WW + x) * CC + ci) = acc;
}

// ---------------------------------------------------------------------------
// K3: inverse L2 norms of the 1536 q rows and 1536 k rows (16384 elems each).
// ---------------------------------------------------------------------------
__global__ __launch_bounds__(256) void k_rownorm(
    const float* __restrict__ Q, const float* __restrict__ K,
    float* __restrict__ IQ, float* __restrict__ IK) {
  const int g = blockIdx.x;                      // 0..3071
  const bool isQ = g < (BATCH * ROWS_PER_BATCH);
  const int row  = isQ ? g : g - BATCH * ROWS_PER_BATCH;
  const float* src = (isQ ? Q : K) + (size_t)row * SLEN;
  float s = 0.f;
  for (int i = threadIdx.x * 4; i < SLEN; i += 256 * 4) {
    float4 v = *(const float4*)(src + i);
    s += v.x * v.x + v.y * v.y + v.z * v.z + v.w * v.w;
  }
  #pragma unroll
  for (int off = 16; off > 0; off >>= 1) s += __shfl_down(s, off, 32);
  __shared__ float red[8];
  if ((threadIdx.x & 31) == 0) red[threadIdx.x >> 5] = s;
  __syncthreads();
  if (threadIdx.x == 0) {
    float tot = 0.f;
    #pragma unroll
    for (int w = 0; w < 8; ++w) tot += red[w];
    (isQ ? IQ : IK)[row] = rsqrtf(fmaxf(tot, 1e-12f));
  }
}

// ---------------------------------------------------------------------------
// K0: zero the attn accumulator (split-K uses f32 global atomics).
// ---------------------------------------------------------------------------
__global__ __launch_bounds__(256) void k_zero(float* __restrict__ p, int n) {
  int i = blockIdx.x * 256 + threadIdx.x;
  if (i < n) p[i] = 0.f;
}

// ---------------------------------------------------------------------------
// K4: attn[pair][48][48] += Q[48,chunk] @ K[48,chunk]^T   (split-K, 64 chunks)
// One wave per block; 9 accumulators (3x3 tiles of 16x16); 576 WMMA per wave.
// ---------------------------------------------------------------------------
#define SCHUNKS 64
__global__ __launch_bounds__(32) void k_attn_gemm(
    const float* __restrict__ Q, const float* __restrict__ K,
    float* __restrict__ ATTN) {
  const int chunk = blockIdx.x & (SCHUNKS - 1);
  const int pair  = blockIdx.x >> 6;           // b*4+n
  const int lane  = threadIdx.x;
  const int li    = lane & 15;
  const int koff  = (lane >> 4) << 1;
  const float* Qb = Q + (size_t)pair * CHEAD * SLEN;
  const float* Kb = K + (size_t)pair * CHEAD * SLEN;
  v8f acc[3][3] = {};
  const int s0 = chunk * (SLEN / SCHUNKS);
  for (int s = s0; s < s0 + (SLEN / SCHUNKS); s += 4) {
    v2f a[3], bm[3];
    #pragma unroll
    for (int mt = 0; mt < 3; ++mt) {
      const float* p = Qb + (size_t)(mt * 16 + li) * SLEN + s + koff;
      a[mt].x = p[0]; a[mt].y = p[1];
    }
    #pragma unroll
    for (int nt = 0; nt < 3; ++nt) {
      const float* p = Kb + (size_t)(nt * 16 + li) * SLEN + s + koff;
      bm[nt].x = p[0]; bm[nt].y = p[1];
    }
    #pragma unroll
    for (int mt = 0; mt < 3; ++mt)
      #pragma unroll
      for (int nt = 0; nt < 3; ++nt)
        acc[mt][nt] = wmma_f32_16x16x4(a[mt], bm[nt], acc[mt][nt]);
  }
  float* A = ATTN + (size_t)pair * CHEAD * CHEAD;
  const int rh = (lane >> 4) << 3;
  #pragma unroll
  for (int mt = 0; mt < 3; ++mt)
    for (int nt = 0; nt < 3; ++nt)
      #pragma unroll
      for (int j = 0; j < 8; ++j)
        atomicAdd(&A[(mt * 16 + rh + j) * CHEAD + nt * 16 + li], acc[mt][nt][j]);
}

// ---------------------------------------------------------------------------
// K5: scale by invq[d]*invk[d']*temperature[n], then rowwise softmax (48).
// One thread per (pair,row): 1536 threads total.
// ---------------------------------------------------------------------------
__global__ __launch_bounds__(256) void k_softmax(
    float* __restrict__ ATTN, const float* __restrict__ IQ,
    const float* __restrict__ IK, const float* __restrict__ TEMP) {
  int gid  = blockIdx.x * 256 + threadIdx.x;   // 1536
  int d    = gid % CHEAD;
  int pair = gid / CHEAD;
  int n    = pair & 3;
  int b    = pair >> 2;
  float* row = ATTN + (size_t)pair * CHEAD * CHEAD + (size_t)d * CHEAD;
  const float* ik = IK + b * ROWS_PER_BATCH + n * CHEAD;
  const float iq  = IQ[b * ROWS_PER_BATCH + n * CHEAD + d] * TEMP[n];
  float vals[CHEAD];
  float mx = -1e30f;
  #pragma unroll
  for (int j = 0; j < CHEAD; ++j) {
    vals[j] = row[j] * iq * ik[j];
    mx = fmaxf(mx, vals[j]);
  }
  float s = 0.f;
  #pragma unroll
  for (int j = 0; j < CHEAD; ++j) { vals[j] = expf(vals[j] - mx); s += vals[j]; }
  const float inv = 1.f / s;
  #pragma unroll
  for (int j = 0; j < CHEAD; ++j) row[j] = vals[j] * inv;
}

// ---------------------------------------------------------------------------
// K6: out_pre[pair][48,16384] = attn[48,48] @ V[48,16384]  (one wave = 16x16)
// ---------------------------------------------------------------------------
__global__ __launch_bounds__(256) void k_av_gemm(
    const float* __restrict__ ATTN, const float* __restrict__ V,
    float* __restrict__ OUTP) {
  const int lane  = threadIdx.x & 31;
  const int wave  = threadIdx.x >> 5;
  const int job   = blockIdx.x * 8 + wave;   // 32 pairs * 3 mtiles * 1024 ntiles
  const int ntile = job & 1023;
  const int mt    = (job >> 10) % 3;
  const int pair  = job / 3072;
  const int li    = lane & 15;
  const int koff  = (lane >> 4) << 1;
  const float* A  = ATTN + (size_t)pair * CHEAD * CHEAD + (size_t)(mt * 16 + li) * CHEAD;
  const float* Vb = V + (size_t)pair * CHEAD * SLEN + ntile * 16 + li;
  v8f acc = {};
  #pragma unroll
  for (int k = 0; k < CHEAD; k += 4) {
    v2f a;  a.x  = A[k + koff];                 a.y  = A[k + koff + 1];
    v2f bv; bv.x = Vb[(size_t)(k + koff) * SLEN]; bv.y = Vb[(size_t)(k + koff + 1) * SLEN];
    acc = wmma_f32_16x16x4(a, bv, acc);
  }
  float* out = OUTP + (size_t)pair * CHEAD * SLEN + ntile * 16 + li;
  const int rh = (lane >> 4) << 3;
  #pragma unroll
  for (int j = 0; j < 8; ++j)
    out[(size_t)(mt * 16 + rh + j) * SLEN] = acc[j];
}

// ---------------------------------------------------------------------------
// K7: d_out = out_pre[131072,192] @ proj_w[192,192]  (one wave -> 16x64 tile)
// ---------------------------------------------------------------------------
__global__ __launch_bounds__(256) void k_proj_gemm(
    const float* __restrict__ P, const float* __restrict__ W,
    float* __restrict__ OUT) {
  const int lane = threadIdx.x & 31;
  const int wave = threadIdx.x >> 5;
  const int job  = blockIdx.x * 8 + wave;    // 8192 mtiles * 3 nsupers
  const int mtile = job / 3;
  const int n0    = (job % 3) * 64;
  const int li    = lane & 15;
  const int koff  = (lane >> 4) << 1;
  const float* arow = P + (size_t)(mtile * 16 + li) * CC;
  v8f acc0 = {}, acc1 = {}, acc2 = {}, acc3 = {};
  #pragma unroll 4
  for (int k = 0; k < CC; k += 4) {
    v2f a; a.x = arow[k + koff]; a.y = arow[k + koff + 1];
    const float* b0 = W + (size_t)(k + koff) * CC + n0 + li;
    const float* b1 = b0 + CC;
    v2f b;
    b.x = b0[0];  b.y = b1[0];  acc0 = wmma_f32_16x16x4(a, b, acc0);
    b.x = b0[16]; b.y = b1[16]; acc1 = wmma_f32_16x16x4(a, b, acc1);
    b.x = b0[32]; b.y = b1[32]; acc2 = wmma_f32_16x16x4(a, b, acc2);
    b.x = b0[48]; b.y = b1[48]; acc3 = wmma_f32_16x16x4(a, b, acc3);
  }
  const int rbase = mtile * 16 + ((lane >> 4) << 3);
  float* out = OUT + n0 + li;
  #pragma unroll
  for (int j = 0; j < 8; ++j) {
    size_t r = (size_t)(rbase + j) * CC;
    out[r +  0] = acc0[j];
    out[r + 16] = acc1[j];
    out[r + 32] = acc2[j];
    out[r + 48] = acc3[j];
  }
}

// ---------------------------------------------------------------------------
// Workspace layout (floats):
//   [0)                qkv (B*HW*576)  -- later aliased as out_pre (B*HW*192)
//   [OFF_Q)            q   (B*HW*192)
//   [OFF_K)            k
//   [OFF_V)            v
//   [OFF_IQ)           inv-norm q (1536)
//   [OFF_IK)           inv-norm k (1536)
//   [OFF_ATTN)         attn (32*48*48)
// ---------------------------------------------------------------------------
#define QKV_ELEMS  ((size_t)BATCH * HWSZ * C3)          // 75497472
#define QBUF_ELEMS ((size_t)BATCH * QBUF_PER_BATCH)     // 25165824
#define OFF_Q   (QKV_ELEMS)
#define OFF_K   (OFF_Q + QBUF_ELEMS)
#define OFF_V   (OFF_K + QBUF_ELEMS)
#define OFF_IQ  (OFF_V + QBUF_ELEMS)
#define OFF_IK  (OFF_IQ + (size_t)(BATCH * ROWS_PER_BATCH))
#define OFF_ATTN (OFF_IK + (size_t)(BATCH * ROWS_PER_BATCH))
#define ATTN_ELEMS (PAIRS * CHEAD * CHEAD)              // 73728

extern "C" void kernel_launch(void* const* d_in, const int* in_sizes, int n_in,
                              void* d_out, int out_size, void* d_ws, size_t ws_size,
                              hipStream_t stream) {
  const float* x      = (const float*)d_in[0];
  const float* qkv_w  = (const float*)d_in[1];
  const float* dw_w   = (const float*)d_in[2];
  const float* proj_w = (const float*)d_in[3];
  const float* temp   = (const float*)d_in[4];

  float* ws    = (float*)d_ws;
  float* qkv   = ws;
  float* qbuf  = ws + OFF_Q;
  float* kbuf  = ws + OFF_K;
  float* vbuf  = ws + OFF_V;
  float* iq    = ws + OFF_IQ;
  float* ik    = ws + OFF_IK;
  float* attn  = ws + OFF_ATTN;
  float* outp  = ws;          // alias qkv region (dead after depthwise)
  float* out   = (float*)d_out;

  // 1) qkv = x @ qkv_w
  k_qkv_gemm<<<9216, 256, 0, stream>>>(x, qkv_w, qkv);
  // 2) depthwise 3x3 + split into contiguous q/k/v (= raw reshape layout)
  k_dwconv_split<<<73728, 256, 0, stream>>>(qkv, dw_w, qbuf, kbuf, vbuf);
  // 3) inverse L2 row norms of q and k
  k_rownorm<<<3072, 256, 0, stream>>>(qbuf, kbuf, iq, ik);
  // 4) attn = q @ k^T   (split-K atomics; zero first)
  k_zero<<<(ATTN_ELEMS + 255) / 256, 256, 0, stream>>>(attn, ATTN_ELEMS);
  k_attn_gemm<<<PAIRS * SCHUNKS, 32, 0, stream>>>(qbuf, kbuf, attn);
  // 5) fold normalization + temperature, softmax
  k_softmax<<<6, 256, 0, stream>>>(attn, iq, ik, temp);
  // 6) out_pre = attn @ v
  k_av_gemm<<<12288, 256, 0, stream>>>(attn, vbuf, outp);
  // 7) out = out_pre @ proj_w
  k_proj_gemm<<<3072, 256, 0, stream>>>(outp, proj_w, out);
}